// CMSBlockLinear_67216238182784
// MI455X (gfx1250) — compile-verified
//
#include <hip/hip_runtime.h>

typedef float v2f __attribute__((ext_vector_type(2)));
typedef float v8f __attribute__((ext_vector_type(8)));

constexpr int IN_F   = 2048;
constexpr int OUT_F  = 2048;
constexpr int TILE_B = 16;
constexpr int KB     = 32;    // selected blocks per output row
constexpr int TOKENS = 2048;  // 4 * 512
constexpr int WAVES_PER_BLOCK = 8;

__global__ __launch_bounds__(256)
void cms_block_linear_wmma(const float* __restrict__ x,
                           const float* __restrict__ values,
                           const int*   __restrict__ cols,
                           const float* __restrict__ bias,
                           float*       __restrict__ out)
{
    const int lane  = threadIdx.x & 31;
    const int wave  = threadIdx.x >> 5;
    const int r     = blockIdx.y;                          // output block row: 0..127
    const int ttile = blockIdx.x * WAVES_PER_BLOCK + wave; // token tile: 0..127
    const int t0    = ttile * TILE_B;

    const int lm = lane & 15;   // A: token row within tile / B,D: output col
    const int lh = lane >> 4;   // selects K-pair {0,1} vs {2,3} within each 4-chunk

    const float* xrow  = x + (size_t)(t0 + lm) * IN_F;              // this lane's token row
    const int*   crow  = cols + r * KB;                              // wave-uniform -> s_load
    const float* vbase = values + (size_t)r * KB * TILE_B * TILE_B;  // r's weight blocks

    v8f acc0 = {}, acc1 = {}, acc2 = {}, acc3 = {};

    #pragma unroll 2
    for (int k = 0; k < KB; ++k) {
        const int c = crow[k];                                   // scalar (uniform) load
        const float* xa = xrow + c * TILE_B + 2 * lh;            // A: x[t, c*16 + j]
        const float* vb = vbase + (size_t)(k * TILE_B + lm) * TILE_B + 2 * lh; // B: values[r,k,i,j]

        v2f a0 = *(const v2f*)(xa + 0);
        v2f a1 = *(const v2f*)(xa + 4);
        v2f a2 = *(const v2f*)(xa + 8);
        v2f a3 = *(const v2f*)(xa + 12);
        v2f b0 = *(const v2f*)(vb + 0);
        v2f b1 = *(const v2f*)(vb + 4);
        v2f b2 = *(const v2f*)(vb + 8);
        v2f b3 = *(const v2f*)(vb + 12);

        // D[t,i] += A(16x4) * B(4x16); 4 independent accumulators break the RAW chain
        acc0 = __builtin_amdgcn_wmma_f32_16x16x4_f32(false, a0, false, b0, (short)0, acc0, false, false);
        acc1 = __builtin_amdgcn_wmma_f32_16x16x4_f32(false, a1, false, b1, (short)0, acc1, false, false);
        acc2 = __builtin_amdgcn_wmma_f32_16x16x4_f32(false, a2, false, b2, (short)0, acc2, false, false);
        acc3 = __builtin_amdgcn_wmma_f32_16x16x4_f32(false, a3, false, b3, (short)0, acc3, false, false);
    }

    v8f d = (acc0 + acc1) + (acc2 + acc3);
    const float bv = bias[r * TILE_B + lm];

    // D layout: VGPR v -> token row (v + 8*lh), output col lm
    #pragma unroll
    for (int v = 0; v < 8; ++v) {
        const int t = t0 + v + 8 * lh;
        out[(size_t)t * OUT_F + r * TILE_B + lm] = d[v] + bv;
    }
}

extern "C" void kernel_launch(void* const* d_in, const int* in_sizes, int n_in,
                              void* d_out, int out_size, void* d_ws, size_t ws_size,
                              hipStream_t stream) {
    const float* x      = (const float*)d_in[0];
    const float* values = (const float*)d_in[1];
    const int*   cols   = (const int*)  d_in[2];
    const float* bias   = (const float*)d_in[3];
    float*       out    = (float*)d_out;

    dim3 grid(TOKENS / TILE_B / WAVES_PER_BLOCK, OUT_F / TILE_B); // (16, 128)
    dim3 block(256);
    hipLaunchKernelGGL(cms_block_linear_wmma, grid, block, 0, stream,
                       x, values, cols, bias, out);
}